// MultiHeadAttention_31679678775480
// MI455X (gfx1250) — compile-verified
//
#include <hip/hip_runtime.h>
#include <hip/hip_bf16.h>

// Problem constants (from reference)
#define B_  16
#define T_  2048
#define D_  256
#define H_  8
#define DK_ 256
#define HD_ (H_ * DK_)          // 2048
#define SCALE_ 0.0625f          // 256^-0.5

typedef float v2f __attribute__((ext_vector_type(2)));
typedef float v8f __attribute__((ext_vector_type(8)));

// ---------------------------------------------------------------------------
// Kernel 1: x_mean[b,d] = mean_t x[b,t,d]     (grid=B, block=D)
// ---------------------------------------------------------------------------
__global__ void k_xmean(const float* __restrict__ x, float* __restrict__ xmean) {
    int b = blockIdx.x;
    int d = threadIdx.x;
    const float* xb = x + (size_t)b * T_ * D_ + d;
    float s = 0.0f;
    for (int t = 0; t < T_; ++t) s += xb[(size_t)t * D_];
    xmean[b * D_ + d] = s * (1.0f / T_);
}

// ---------------------------------------------------------------------------
// Kernel 2: per (b,h):
//   q_mean   = x_mean @ Wq_h + bq_h
//   q_master = q_mean @ Wm + bm
//   wtil[i]  = sum_j Wk[i, h*DK+j] * q_master[j]
//   cbh      = sum_j bk[h*DK+j]   * q_master[j]
// grid = B*H, block = 256
// ---------------------------------------------------------------------------
__global__ void k_proj(const float* __restrict__ xmean,
                       const float* __restrict__ Wq, const float* __restrict__ bq,
                       const float* __restrict__ Wm, const float* __restrict__ bm,
                       const float* __restrict__ Wk, const float* __restrict__ bk,
                       float* __restrict__ wtil, float* __restrict__ cbh) {
    int bh = blockIdx.x;
    int b = bh / H_, h = bh % H_;
    int tid = threadIdx.x;

    __shared__ float sxm[D_];
    __shared__ float sqm[DK_];
    __shared__ float sqmast[DK_];
    __shared__ float sred[256];

    sxm[tid] = xmean[b * D_ + tid];
    __syncthreads();

    // q_mean[tid]
    {
        float acc = bq[h * DK_ + tid];
        for (int i = 0; i < D_; ++i)
            acc += sxm[i] * Wq[(size_t)i * HD_ + h * DK_ + tid];
        sqm[tid] = acc;
    }
    __syncthreads();

    // q_master[tid]
    {
        float acc = bm[tid];
        for (int i = 0; i < DK_; ++i)
            acc += sqm[i] * Wm[(size_t)i * DK_ + tid];
        sqmast[tid] = acc;
    }
    __syncthreads();

    // wtil[i = tid]
    {
        float acc = 0.0f;
        const float* wkrow = Wk + (size_t)tid * HD_ + h * DK_;
        for (int j = 0; j < DK_; ++j) acc += wkrow[j] * sqmast[j];
        wtil[((size_t)b * H_ + h) * D_ + tid] = acc;
    }

    // cbh = dot(bk_h, q_master)
    sred[tid] = bk[h * DK_ + tid] * sqmast[tid];
    __syncthreads();
    for (int s = 128; s > 0; s >>= 1) {
        if (tid < s) sred[tid] += sred[tid + s];
        __syncthreads();
    }
    if (tid == 0) cbh[b * H_ + h] = sred[0];
}

// ---------------------------------------------------------------------------
// Kernel 3: raw scores via WMMA f32 16x16x4.
//   scores[b,h,t] = SCALE * ( x[b,t,:] . wtil[b,h,:] + cbh[b,h] )
// A tile = 16 t-rows x 4 K of x;  B tile = wtil^T (D x 16, heads padded to 16).
// grid = (B, 8)  (8 tile-groups of 16 tiles), block = 256 (8 waves).
// Each wave computes 2 output tiles of 16 t x 16 cols (cols 0..7 = heads).
// ---------------------------------------------------------------------------
__global__ void k_scores(const float* __restrict__ x,
                         const float* __restrict__ wtil,
                         const float* __restrict__ cbh,
                         float* __restrict__ attn /* [B,H,T] */) {
    int b = blockIdx.x;
    int tid = threadIdx.x;
    int wave = tid >> 5, lane = tid & 31;
    int hi = lane >> 4;        // K-pair select
    int lm = lane & 15;        // M (A rows) or N (B/D cols)

    __shared__ float sB[D_][16];   // wtil transposed, padded: 16 KB
    __shared__ float sC[H_];

    for (int idx = tid; idx < D_ * 16; idx += 256) {
        int k = idx >> 4, n = idx & 15;
        sB[k][n] = (n < H_) ? wtil[((size_t)b * H_ + n) * D_ + k] : 0.0f;
    }
    if (tid < H_) sC[tid] = cbh[b * H_ + tid];
    __syncthreads();

    // tiles handled by this (blockIdx.y, wave): 2 tiles each, 128 total
    for (int sub = 0; sub < 2; ++sub) {
        int tile = blockIdx.y * 16 + wave * 2 + sub;
        int t0 = tile * 16;
        const float* xrow = x + ((size_t)b * T_ + t0 + lm) * D_;

        v8f c = {};
        for (int k0 = 0; k0 < D_; k0 += 4) {
            int ka = k0 + 2 * hi;
            v2f a;  a.x  = xrow[ka];      a.y  = xrow[ka + 1];
            v2f bb; bb.x = sB[ka][lm];    bb.y = sB[ka + 1][lm];
            c = __builtin_amdgcn_wmma_f32_16x16x4_f32(
                    false, a, false, bb, (short)0, c, false, false);
        }

        if (lm < H_) {                     // valid head columns only
            float cc = sC[lm];
            float* srow = attn + ((size_t)b * H_ + lm) * T_;
#pragma unroll
            for (int r = 0; r < 8; ++r)
                srow[t0 + r + 8 * hi] = (c[r] + cc) * SCALE_;
        }
    }
}

// ---------------------------------------------------------------------------
// Kernel 4: softmax over t, in place in attn.  grid = B*H, block = 256.
// ---------------------------------------------------------------------------
__global__ void k_softmax(float* __restrict__ attn) {
    int bh = blockIdx.x;
    int tid = threadIdx.x;
    float* row = attn + (size_t)bh * T_;
    __shared__ float red[256];

    float m = -3.4e38f;
    for (int t = tid; t < T_; t += 256) m = fmaxf(m, row[t]);
    red[tid] = m; __syncthreads();
    for (int s = 128; s > 0; s >>= 1) {
        if (tid < s) red[tid] = fmaxf(red[tid], red[tid + s]);
        __syncthreads();
    }
    m = red[0]; __syncthreads();

    float sum = 0.0f;
    for (int t = tid; t < T_; t += 256) {
        float e = __expf(row[t] - m);
        row[t] = e;
        sum += e;
    }
    red[tid] = sum; __syncthreads();
    for (int s = 128; s > 0; s >>= 1) {
        if (tid < s) red[tid] += red[tid + s];
        __syncthreads();
    }
    float inv = 1.0f / red[0];
    __syncthreads();
    for (int t = tid; t < T_; t += 256) row[t] *= inv;
}

// ---------------------------------------------------------------------------
// Kernel 5: u[b,h,d] = sum_t attn[b,h,t] * x[b,t,d]   via WMMA f32 16x16x4.
// A = attn rows (heads 0..7, padded to 16), B = x[t, d0..d0+15].
// grid = (B, 2), block = 256 (8 waves), wave -> one 16-col d tile.
// Only lanes 0-15 (D rows M=0..7 => heads) write results.
// ---------------------------------------------------------------------------
__global__ void k_u(const float* __restrict__ x,
                    const float* __restrict__ attn,
                    float* __restrict__ u) {
    int b = blockIdx.x;
    int tid = threadIdx.x;
    int wave = tid >> 5, lane = tid & 31;
    int hi = lane >> 4;
    int lm = lane & 15;

    int dt = blockIdx.y * 8 + wave;   // d tile 0..15
    int d0 = dt * 16;

    const float* arow = attn + ((size_t)b * H_ + (lm & 7)) * T_;
    float msk = (lm < H_) ? 1.0f : 0.0f;
    const float* xcol = x + (size_t)b * T_ * D_ + d0 + lm;

    v8f c = {};
    for (int k0 = 0; k0 < T_; k0 += 4) {
        int ka = k0 + 2 * hi;
        v2f a;  a.x  = msk * arow[ka];           a.y  = msk * arow[ka + 1];
        v2f bb; bb.x = xcol[(size_t)ka * D_];    bb.y = xcol[(size_t)(ka + 1) * D_];
        c = __builtin_amdgcn_wmma_f32_16x16x4_f32(
                false, a, false, bb, (short)0, c, false, false);
    }

    if (hi == 0) {   // lanes 0-15 hold M = r = head 0..7
#pragma unroll
        for (int r = 0; r < 8; ++r)
            u[((size_t)b * H_ + r) * D_ + d0 + lm] = c[r];
    }
}

// ---------------------------------------------------------------------------
// Kernel 6: epilogue.
//   out_h = u @ Wv + bv          [B,H,D]
//   out[b,j] = sum_{d,h} out_h[b,h,d] * Wf[(d*H+h)*D + j] + bf[j]
// grid = B, block = 256.
// ---------------------------------------------------------------------------
__global__ void k_final(const float* __restrict__ u,
                        const float* __restrict__ Wv, const float* __restrict__ bv,
                        const float* __restrict__ Wf, const float* __restrict__ bf,
                        float* __restrict__ out) {
    int b = blockIdx.x;
    int tid = threadIdx.x;
    __shared__ float sU[H_][D_];    // 8 KB
    __shared__ float sOH[H_][D_];   // 8 KB

    for (int idx = tid; idx < H_ * D_; idx += 256)
        sU[idx >> 8][idx & 255] = u[(size_t)b * H_ * D_ + idx];
    __syncthreads();

    for (int h = 0; h < H_; ++h) {
        float acc = bv[tid];
        for (int i = 0; i < D_; ++i)
            acc += sU[h][i] * Wv[(size_t)i * D_ + tid];
        sOH[h][tid] = acc;
    }
    __syncthreads();

    float acc = bf[tid];
    for (int d = 0; d < D_; ++d) {
#pragma unroll
        for (int h = 0; h < H_; ++h)
            acc += sOH[h][d] * Wf[((size_t)d * H_ + h) * D_ + tid];
    }
    out[b * D_ + tid] = acc;
}

// ---------------------------------------------------------------------------
extern "C" void kernel_launch(void* const* d_in, const int* in_sizes, int n_in,
                              void* d_out, int out_size, void* d_ws, size_t ws_size,
                              hipStream_t stream) {
    const float* x  = (const float*)d_in[0];
    const float* Wk = (const float*)d_in[1];
    const float* bk = (const float*)d_in[2];
    const float* Wq = (const float*)d_in[3];
    const float* bq = (const float*)d_in[4];
    const float* Wv = (const float*)d_in[5];
    const float* bv = (const float*)d_in[6];
    const float* Wm = (const float*)d_in[7];
    const float* bm = (const float*)d_in[8];
    const float* Wf = (const float*)d_in[9];
    const float* bf = (const float*)d_in[10];

    float* out  = (float*)d_out;            // [B,D] = 4096 floats
    float* attn = out + B_ * D_;            // [B,H,T] = 262144 floats

    // workspace layout (floats): xmean | wtil | cbh | u  => ~279 KB total
    float* ws    = (float*)d_ws;
    float* xmean = ws;                                  // B*D      = 4096
    float* wtil  = xmean + B_ * D_;                     // B*H*D    = 32768
    float* cbh   = wtil + B_ * H_ * D_;                 // B*H      = 128
    float* u     = cbh + B_ * H_;                       // B*H*D    = 32768

    k_xmean  <<<B_,            256, 0, stream>>>(x, xmean);
    k_proj   <<<B_ * H_,       256, 0, stream>>>(xmean, Wq, bq, Wm, bm, Wk, bk, wtil, cbh);
    k_scores <<<dim3(B_, 8),   256, 0, stream>>>(x, wtil, cbh, attn);
    k_softmax<<<B_ * H_,       256, 0, stream>>>(attn);
    k_u      <<<dim3(B_, 2),   256, 0, stream>>>(x, attn, u);
    k_final  <<<B_,            256, 0, stream>>>(u, Wv, bv, Wf, bf, out);
}